// MultiBoxLoss_13855564497184
// MI455X (gfx1250) — compile-verified
//
#include <hip/hip_runtime.h>
#include <hip/hip_bf16.h>
#include <math.h>

#define BSZ   128
#define DBOX  8732
#define NCLS  21
#define NOBJ  16
#define TPB   256
#define NWAVE (TPB / 32)
#define NTILE ((DBOX + TPB - 1) / TPB)   // 35

// ---------------- CDNA5 async global->LDS path (gfx1250) ----------------
#if defined(__has_builtin)
#if __has_builtin(__builtin_amdgcn_global_load_async_to_lds_b32)
#define HAVE_ASYNC_B32 1
#endif
#if __has_builtin(__builtin_amdgcn_s_wait_asynccnt)
#define HAVE_WAIT_ASYNC 1
#endif
#endif

typedef __attribute__((address_space(1))) int as1_int;
typedef __attribute__((address_space(3))) int as3_int;

__device__ __forceinline__ void async_ld_f32(const float* g, float* l) {
#if defined(HAVE_ASYNC_B32)
  __builtin_amdgcn_global_load_async_to_lds_b32(
      (as1_int*)(void*)g,
      (as3_int*)l,
      0, 0);
#else
  *l = *g;
#endif
}

// wait until at most the NCLS async loads of the in-flight tile remain
__device__ __forceinline__ void async_wait_keep_tile() {
#if defined(HAVE_ASYNC_B32)
#if defined(HAVE_WAIT_ASYNC)
  __builtin_amdgcn_s_wait_asynccnt(NCLS);   // in-order completion => prev tile done
#else
  asm volatile("s_wait_asynccnt 21" ::: "memory");
#endif
#endif
}

__device__ __forceinline__ void async_wait_all() {
#if defined(HAVE_ASYNC_B32)
#if defined(HAVE_WAIT_ASYNC)
  __builtin_amdgcn_s_wait_asynccnt(0);
#else
  asm volatile("s_wait_asynccnt 0" ::: "memory");
#endif
#endif
}

// ---------------- block reductions (deterministic, wave32) ----------------
__device__ __forceinline__ float block_sum_f(float v, float* sbuf) {
  int lane = threadIdx.x & 31, wv = threadIdx.x >> 5;
#pragma unroll
  for (int off = 16; off > 0; off >>= 1) v += __shfl_down(v, off, 32);
  __syncthreads();
  if (lane == 0) sbuf[wv] = v;
  __syncthreads();
  if (threadIdx.x == 0) {
    float r = 0.f;
    for (int i = 0; i < NWAVE; i++) r += sbuf[i];
    sbuf[0] = r;
  }
  __syncthreads();
  return sbuf[0];
}

__device__ __forceinline__ int block_sum_i(int v, int* sbuf) {
  int lane = threadIdx.x & 31, wv = threadIdx.x >> 5;
#pragma unroll
  for (int off = 16; off > 0; off >>= 1) v += __shfl_down(v, off, 32);
  __syncthreads();
  if (lane == 0) sbuf[wv] = v;
  __syncthreads();
  if (threadIdx.x == 0) {
    int r = 0;
    for (int i = 0; i < NWAVE; i++) r += sbuf[i];
    sbuf[0] = r;
  }
  __syncthreads();
  return sbuf[0];
}

// ---------------- kernel 1: SSD matching ----------------
__global__ __launch_bounds__(TPB) void match_kernel(
    const float* __restrict__ gt_boxes, const float* __restrict__ dboxes,
    float* __restrict__ ovl, int* __restrict__ bidx) {
  const int b = blockIdx.x;
  const int tid = threadIdx.x;
  __shared__ float gx1[NOBJ], gy1[NOBJ], gx2[NOBJ], gy2[NOBJ], gar[NOBJ];
  __shared__ float s_bv[NOBJ * NWAVE];
  __shared__ int s_bi[NOBJ * NWAVE];
  __shared__ int s_bd[NOBJ];

  if (tid < NOBJ) {
    const float* gb = gt_boxes + ((size_t)b * NOBJ + tid) * 4;
    float x1 = gb[0], y1 = gb[1], x2 = gb[2], y2 = gb[3];
    gx1[tid] = x1; gy1[tid] = y1; gx2[tid] = x2; gy2[tid] = y2;
    gar[tid] = (x2 - x1) * (y2 - y1);
  }
  __syncthreads();

  float tv[NOBJ];
  int tix[NOBJ];
#pragma unroll
  for (int o = 0; o < NOBJ; o++) { tv[o] = -1.0f; tix[o] = 0x7fffffff; }

  for (int d = tid; d < DBOX; d += TPB) {
    const float4 db = ((const float4*)dboxes)[d];
    float hw = db.z * 0.5f, hh = db.w * 0.5f;
    float dx1 = db.x - hw, dy1 = db.y - hh, dx2 = db.x + hw, dy2 = db.y + hh;
    float dar = db.z * db.w;
    float bv = -1.0f;
    int bo = 0;
#pragma unroll
    for (int o = 0; o < NOBJ; o++) {
      float ix1 = fmaxf(gx1[o], dx1), iy1 = fmaxf(gy1[o], dy1);
      float ix2 = fminf(gx2[o], dx2), iy2 = fminf(gy2[o], dy2);
      float iw = fmaxf(ix2 - ix1, 0.f), ih = fmaxf(iy2 - iy1, 0.f);
      float inter = iw * ih;
      float iou = inter / (gar[o] + dar - inter);
      if (iou > bv) { bv = iou; bo = o; }  // first-max == jnp.argmax(axis=0)
      if (iou > tv[o] || (iou == tv[o] && d < tix[o])) { tv[o] = iou; tix[o] = d; }
    }
    ovl[(size_t)b * DBOX + d] = bv;
    bidx[(size_t)b * DBOX + d] = bo;
  }

  // per-truth argmax over D (smallest index on ties == jnp.argmax(axis=1))
  int lane = tid & 31, wv = tid >> 5;
#pragma unroll
  for (int o = 0; o < NOBJ; o++) {
    float v = tv[o];
    int ix = tix[o];
    for (int off = 16; off > 0; off >>= 1) {
      float v2 = __shfl_down(v, off, 32);
      int i2 = __shfl_down(ix, off, 32);
      if (v2 > v || (v2 == v && i2 < ix)) { v = v2; ix = i2; }
    }
    if (lane == 0) { s_bv[o * NWAVE + wv] = v; s_bi[o * NWAVE + wv] = ix; }
  }
  __syncthreads();
  if (tid < NOBJ) {
    float v = s_bv[tid * NWAVE];
    int ix = s_bi[tid * NWAVE];
    for (int w = 1; w < NWAVE; w++) {
      float v2 = s_bv[tid * NWAVE + w];
      int i2 = s_bi[tid * NWAVE + w];
      if (v2 > v || (v2 == v && i2 < ix)) { v = v2; ix = i2; }
    }
    s_bd[tid] = ix;
  }
  __syncthreads();
  if (tid == 0) {  // sequential override: bt_idx[best_default[o]] = o, overlap = 1
    for (int o = 0; o < NOBJ; o++) {
      int d = s_bd[o];
      bidx[(size_t)b * DBOX + d] = o;
      ovl[(size_t)b * DBOX + d] = 1.0f;
    }
  }
}

// ---------------- kernel 2: CE + loc loss + hard-negative top-K ----------------
// Double-buffered async staging of cls_pred tiles into LDS. Each thread issues
// exactly NCLS async loads per tile (tail indices clamped to 0 so every wave's
// ASYNCcnt rises by exactly NCLS per tile); async loads complete in order, so
// s_wait_asynccnt(NCLS) guarantees the previous tile has fully landed.
__global__ __launch_bounds__(TPB) void loss_kernel(
    const float* __restrict__ loc_pred, const float* __restrict__ cls_pred,
    const float* __restrict__ gt_boxes, const int* __restrict__ gt_labels,
    const float* __restrict__ dboxes, const float* __restrict__ ovl,
    const int* __restrict__ bidx, int* __restrict__ nposB,
    float* __restrict__ locB, float* __restrict__ posB, float* __restrict__ negB) {
  const int b = blockIdx.x;
  const int tid = threadIdx.x;
  __shared__ float negce[DBOX];               // 34928 B: negative CE, LDS-resident
  __shared__ float tiles[2][TPB * NCLS];      // 2x 21504 B: async-staged cls_pred
  __shared__ float s_f[NWAVE];
  __shared__ int s_i[NWAVE];
  __shared__ float gbx[NOBJ * 4];
  __shared__ int glab[NOBJ];

  const float* base = cls_pred + (size_t)b * DBOX * NCLS;

  // issue tile 0 before anything else so it streams during setup
  {
    int total0 = min(TPB, DBOX) * NCLS;
#pragma unroll
    for (int k = 0; k < NCLS; k++) {
      int i = tid + k * TPB;
      int ii = (i < total0) ? i : 0;
      async_ld_f32(base + ii, &tiles[0][ii]);
    }
  }

  if (tid < NOBJ * 4) gbx[tid] = gt_boxes[(size_t)b * NOBJ * 4 + tid];
  if (tid < NOBJ) glab[tid] = gt_labels[(size_t)b * NOBJ + tid];

  float pos_acc = 0.f, loc_acc = 0.f;
  int npos_loc = 0;

  for (int t = 0; t < NTILE; t++) {
    int t0 = t * TPB;
    int n = min(TPB, DBOX - t0);

    // prefetch next tile into the other buffer
    if (t + 1 < NTILE) {
      int nt0 = (t + 1) * TPB;
      int ntotal = min(TPB, DBOX - nt0) * NCLS;
      const float* nsrc = base + (size_t)nt0 * NCLS;
      float* nbuf = tiles[(t + 1) & 1];
#pragma unroll
      for (int k = 0; k < NCLS; k++) {
        int i = tid + k * TPB;
        int ii = (i < ntotal) ? i : 0;
        async_ld_f32(nsrc + ii, &nbuf[ii]);
      }
      async_wait_keep_tile();   // <= NCLS outstanding => tile t complete
    } else {
      async_wait_all();
    }
    __syncthreads();

    if (tid < n) {
      int d = t0 + tid;
      const float* row = &tiles[t & 1][tid * NCLS];  // stride 21: conflict-free
      float m = row[0];
#pragma unroll
      for (int c = 1; c < NCLS; c++) m = fmaxf(m, row[c]);
      float s = 0.f;
#pragma unroll
      for (int c = 0; c < NCLS; c++) s += expf(row[c] - m);
      float lse = m + logf(s);
      float ov = ovl[(size_t)b * DBOX + d];
      int ix = bidx[(size_t)b * DBOX + d];
      int lab = (ov < 0.5f) ? 0 : glab[ix];
      float ce = lse - row[lab];
      if (lab > 0) {
        pos_acc += ce;
        npos_loc++;
        // encode matched gt (xyxy->cxcywh) vs prior (cxcywh), variances (0.1,0.2)
        float bx1 = gbx[ix * 4 + 0], by1 = gbx[ix * 4 + 1];
        float bx2 = gbx[ix * 4 + 2], by2 = gbx[ix * 4 + 3];
        float cx = (bx1 + bx2) * 0.5f, cy = (by1 + by2) * 0.5f;
        float w = bx2 - bx1, h = by2 - by1;
        const float4 p = ((const float4*)dboxes)[d];
        float tgt[4];
        tgt[0] = (cx - p.x) / (p.z / 10.0f);
        tgt[1] = (cy - p.y) / (p.w / 10.0f);
        tgt[2] = logf(w / p.z) * 5.0f;
        tgt[3] = logf(h / p.w) * 5.0f;
        const float* lp = loc_pred + ((size_t)b * DBOX + d) * 4;
#pragma unroll
        for (int k = 0; k < 4; k++) {
          float df = lp[k] - tgt[k];
          float ad = fabsf(df);
          loc_acc += (ad < 1.0f) ? 0.5f * df * df : (ad - 0.5f);
        }
        negce[d] = 0.f;
      } else {
        negce[d] = ce;
      }
    }
    __syncthreads();  // buffer t&1 free for reuse at t+2
  }

  float pos_b = block_sum_f(pos_acc, s_f);
  float loc_b = block_sum_f(loc_acc, s_f);
  int np_b = block_sum_i(npos_loc, s_i);

  int K = 3 * np_b;
  if (K > DBOX) K = DBOX;

  // exact top-K sum via bitwise radix-select (CE >= 0 so uint order == float order)
  float neg_b = 0.f;
  unsigned pref = 0u;
  if (K > 0) {
    for (int bit = 31; bit >= 0; --bit) {
      unsigned cand = pref | (1u << bit);
      int c = 0;
      for (int i = tid; i < DBOX; i += TPB)
        c += (__float_as_uint(negce[i]) >= cand) ? 1 : 0;
      c = block_sum_i(c, s_i);
      if (c >= K) pref = cand;  // pref -> max t with count(x>=t) >= K == K-th largest
    }
    float thr = __uint_as_float(pref);
    float sgt = 0.f;
    int cgt = 0;
    for (int i = tid; i < DBOX; i += TPB) {
      float x = negce[i];
      if (__float_as_uint(x) > pref) { sgt += x; cgt++; }
    }
    sgt = block_sum_f(sgt, s_f);
    cgt = block_sum_i(cgt, s_i);
    neg_b = sgt + (float)(K - cgt) * thr;  // ties at threshold counted exactly
  }

  if (tid == 0) {
    nposB[b] = np_b;
    locB[b] = loc_b;
    posB[b] = pos_b;
    negB[b] = neg_b;
  }
}

// ---------------- kernel 3: deterministic final combine ----------------
__global__ void finalize_kernel(const int* __restrict__ nposB,
                                const float* __restrict__ locB,
                                const float* __restrict__ posB,
                                const float* __restrict__ negB,
                                float* __restrict__ out, int B) {
  if (threadIdx.x == 0 && blockIdx.x == 0) {
    long long npi = 0;
    float lp = 0.f, pp = 0.f, nn = 0.f;
    for (int b = 0; b < B; b++) {
      npi += nposB[b];
      lp += locB[b];
      pp += posB[b];
      nn += negB[b];
    }
    float N = (float)npi;
    out[0] = lp / (N * 4.0f) + (pp + nn) / N;
  }
}

extern "C" void kernel_launch(void* const* d_in, const int* in_sizes, int n_in,
                              void* d_out, int out_size, void* d_ws, size_t ws_size,
                              hipStream_t stream) {
  const float* loc_pred = (const float*)d_in[0];
  const float* cls_pred = (const float*)d_in[1];
  const float* gt_boxes = (const float*)d_in[2];
  const int* gt_labels = (const int*)d_in[3];
  const float* dboxes = (const float*)d_in[4];
  const int B = in_sizes[3] / NOBJ;  // 128

  // workspace layout (~9 MB)
  float* ovl = (float*)d_ws;                       // [B, DBOX]
  int* bidx = (int*)(ovl + (size_t)B * DBOX);      // [B, DBOX]
  int* nposB = (int*)(bidx + (size_t)B * DBOX);    // [B]
  float* locB = (float*)(nposB + B);               // [B]
  float* posB = locB + B;                          // [B]
  float* negB = posB + B;                          // [B]

  match_kernel<<<B, TPB, 0, stream>>>(gt_boxes, dboxes, ovl, bidx);
  loss_kernel<<<B, TPB, 0, stream>>>(loc_pred, cls_pred, gt_boxes, gt_labels,
                                     dboxes, ovl, bidx, nposB, locB, posB, negB);
  finalize_kernel<<<1, 32, 0, stream>>>(nposB, locB, posB, negB, (float*)d_out, B);
}